// HypergraphConvDHG_33973191311518
// MI455X (gfx1250) — compile-verified
//
#include <hip/hip_runtime.h>
#include <hip/hip_bf16.h>

// ---------------------------------------------------------------------------
// Problem constants (from reference)
// ---------------------------------------------------------------------------
#define BB    8
#define NN    4096
#define EE    1000
#define DEG   8
#define CC    64
#define ROWS  (BB * EE)               // 8000 rows of hef
#define NTM   (ROWS / 64)             // 125 macro row-tiles (64 rows each)
#define TRIM  (NTM * (NTM + 1) / 2)   // 7875 triangular macro jobs
#define LPAD  66                      // LDS row stride (floats), bank-conflict pad
#define MARGIN 4.2f

typedef float v2f __attribute__((ext_vector_type(2)));
typedef float v8f __attribute__((ext_vector_type(8)));

static __device__ __forceinline__ float waveReduceAdd(float v) {
#pragma unroll
  for (int m = 16; m > 0; m >>= 1) v += __shfl_xor(v, m, 32);
  return v;
}

// ---------------------------------------------------------------------------
// K0: zero deg counts + loss accumulators
// ---------------------------------------------------------------------------
__global__ void k_zero(int* __restrict__ deg, float* __restrict__ accum) {
  int i = blockIdx.x * blockDim.x + threadIdx.x;
  if (i < NN) deg[i] = 0;
  if (i < 2) accum[i] = 0.0f;
}

// ---------------------------------------------------------------------------
// K1: per-node degree counts
// ---------------------------------------------------------------------------
__global__ void k_degcount(const int* __restrict__ ids, int* __restrict__ deg) {
  int i = blockIdx.x * blockDim.x + threadIdx.x;
  if (i < EE * DEG) atomicAdd(&deg[ids[i]], 1);
}

// ---------------------------------------------------------------------------
// K2: serial exclusive scan (N=4096, negligible) -> csr offsets + cursors
// ---------------------------------------------------------------------------
__global__ void k_scan(const int* __restrict__ deg, int* __restrict__ off,
                       int* __restrict__ cur) {
  if (threadIdx.x == 0 && blockIdx.x == 0) {
    int run = 0;
    for (int n = 0; n < NN; ++n) { off[n] = run; cur[n] = run; run += deg[n]; }
  }
}

// ---------------------------------------------------------------------------
// K3: fill CSR entries (node -> incident edge occurrence list)
// ---------------------------------------------------------------------------
__global__ void k_fill(const int* __restrict__ ids, int* __restrict__ cur,
                       int* __restrict__ csr_e) {
  int i = blockIdx.x * blockDim.x + threadIdx.x;
  if (i < EE * DEG) {
    int n = ids[i];
    int slot = atomicAdd(&cur[n], 1);
    csr_e[slot] = i >> 3;   // edge id
  }
}

// ---------------------------------------------------------------------------
// K4: C(M,64) = A(M,64) @ W(64,64) * alpha + bias   via V_WMMA_F32_16X16X4_F32
// One wave per 16x16 output tile, 16 WMMA steps over K=64.
// ---------------------------------------------------------------------------
__global__ void k_gemm64(const float* __restrict__ A, const float* __restrict__ W,
                         const float* __restrict__ bias, float* __restrict__ C,
                         int M, float alpha) {
  int wave = (blockIdx.x * blockDim.x + threadIdx.x) >> 5;
  int lane = threadIdx.x & 31;
  int mtiles = M >> 4;
  if (wave >= mtiles * 4) return;            // uniform per wave -> EXEC all ones
  int row0 = (wave >> 2) << 4;
  int n0   = (wave & 3) << 4;
  int m    = lane & 15;
  int half = lane >> 4;

  v8f acc = {0.f, 0.f, 0.f, 0.f, 0.f, 0.f, 0.f, 0.f};
  const float* arow = A + (size_t)(row0 + m) * CC;
#pragma unroll
  for (int kk = 0; kk < 16; ++kk) {
    int kb = kk * 4 + half * 2;
    v2f a = *(const v2f*)(arow + kb);                 // A[m][kb..kb+1]
    v2f b;
    b.x = W[kb * CC + n0 + m];                        // B[kb][n]
    b.y = W[(kb + 1) * CC + n0 + m];                  // B[kb+1][n]
    acc = __builtin_amdgcn_wmma_f32_16x16x4_f32(false, a, false, b,
                                                (short)0, acc, false, false);
  }
  float bv = bias[n0 + m];
#pragma unroll
  for (int v = 0; v < 8; ++v) {
    int row = row0 + v + half * 8;
    C[(size_t)row * CC + n0 + m] = acc[v] * alpha + bv;
  }
}

// ---------------------------------------------------------------------------
// K5: he[b,e,:] = sum_d x_trans[b, ids[e,d], :]
// ---------------------------------------------------------------------------
__global__ void k_he(const float* __restrict__ xt, const int* __restrict__ ids,
                     float* __restrict__ he) {
  int idx = blockIdx.x * blockDim.x + threadIdx.x;
  if (idx >= ROWS * 16) return;
  int c4 = idx & 15;
  int r  = idx >> 4;               // b*E + e
  int e  = r % EE;
  int b  = r / EE;
  float sx = 0.f, sy = 0.f, sz = 0.f, sw = 0.f;
  const int* row = ids + e * DEG;
#pragma unroll
  for (int d = 0; d < DEG; ++d) {
    const float* g = xt + ((size_t)b * NN + row[d]) * CC + c4 * 4;
    sx += g[0]; sy += g[1]; sz += g[2]; sw += g[3];
  }
  float* o = he + (size_t)r * CC + c4 * 4;
  o[0] = sx; o[1] = sy; o[2] = sz; o[3] = sw;
}

// ---------------------------------------------------------------------------
// K6: row squared norms of hef (8000 x 64)
// ---------------------------------------------------------------------------
__global__ void k_sq(const float* __restrict__ hef, float* __restrict__ sq) {
  int r = blockIdx.x * blockDim.x + threadIdx.x;
  if (r >= ROWS) return;
  const float* p = hef + (size_t)r * CC;
  float s = 0.f;
#pragma unroll
  for (int c = 0; c < CC; ++c) s += p[c] * p[c];
  sq[r] = s;
}

// ---------------------------------------------------------------------------
// K7: Gram tiles via WMMA + fused contrastive loss reduction.
// 64x64 macro-tile per block: panels staged in LDS via
// GLOBAL_LOAD_ASYNC_TO_LDS_B128 (ASYNCcnt), 8 waves x two 16x16 WMMA subtiles.
// Triangular macro enumeration; off-diagonal macros weighted 2x by symmetry.
// ---------------------------------------------------------------------------
__global__ void k_gram_loss(const float* __restrict__ hef,
                            const float* __restrict__ sq,
                            float* __restrict__ accum_hyper) {
  __shared__ float lsA[64 * LPAD];
  __shared__ float lsB[64 * LPAD];
  __shared__ float red[8];

  int tid  = threadIdx.x;
  int lane = tid & 31;
  int wib  = tid >> 5;
  int m    = lane & 15;
  int half = lane >> 4;
  float lsum = 0.f;

  for (int job = blockIdx.x; job < TRIM; job += gridDim.x) {
    // decode triangular macro (ti <= tj)
    float ff = (float)((2 * NTM + 1) * (2 * NTM + 1)) - 8.0f * (float)job;
    int ti = (int)(((float)(2 * NTM + 1) - sqrtf(ff)) * 0.5f);
    if (ti < 0) ti = 0;
    if (ti > NTM - 1) ti = NTM - 1;
    while (ti > 0 && (ti * NTM - ti * (ti - 1) / 2) > job) --ti;
    while (((ti + 1) * NTM - (ti + 1) * ti / 2) <= job) ++ti;
    int tj = ti + (job - (ti * NTM - ti * (ti - 1) / 2));
    int i0 = ti * 64, j0 = tj * 64;

    __syncthreads();   // previous job's compute must finish before overwrite

    // stage both 64x64 panels into LDS: 4 b128 slots per thread per panel
#pragma unroll
    for (int p = 0; p < 4; ++p) {
      int slot = tid + p * 256;          // 0..1023
      int row  = slot >> 4;
      int c4   = slot & 15;
      unsigned la = (unsigned)(size_t)&lsA[row * LPAD + c4 * 4];
      const float* ga = hef + (size_t)(i0 + row) * CC + c4 * 4;
      asm volatile("global_load_async_to_lds_b128 %0, %1, off"
                   :: "v"(la), "v"(ga) : "memory");
      unsigned lb = (unsigned)(size_t)&lsB[row * LPAD + c4 * 4];
      const float* gb = hef + (size_t)(j0 + row) * CC + c4 * 4;
      asm volatile("global_load_async_to_lds_b128 %0, %1, off"
                   :: "v"(lb), "v"(gb) : "memory");
    }
    asm volatile("s_wait_asynccnt 0x0" ::: "memory");
    __syncthreads();

    float w = (ti == tj) ? 1.0f : 2.0f;

    // each wave computes two 16x16 subtiles of the 4x4 subtile grid
#pragma unroll
    for (int s = 0; s < 2; ++s) {
      int st = wib * 2 + s;
      int si = st >> 2, sj = st & 3;

      v8f acc = {0.f, 0.f, 0.f, 0.f, 0.f, 0.f, 0.f, 0.f};
      const float* ar = &lsA[(si * 16 + m) * LPAD];
      const float* br = &lsB[(sj * 16 + m) * LPAD];
#pragma unroll
      for (int kk = 0; kk < 16; ++kk) {
        int kb = kk * 4 + half * 2;
        v2f a = *(const v2f*)(ar + kb);
        v2f b = *(const v2f*)(br + kb);
        acc = __builtin_amdgcn_wmma_f32_16x16x4_f32(false, a, false, b,
                                                    (short)0, acc, false, false);
      }

      int j = j0 + sj * 16 + m;
      float sqj = sq[j];
      float nj  = sqrtf(sqj);
#pragma unroll
      for (int v = 0; v < 8; ++v) {
        int i = i0 + si * 16 + v + half * 8;
        float sqi = sq[i];
        float ni  = sqrtf(sqi);
        float g   = acc[v];
        float sim = g / fmaxf(ni * nj, 1e-8f);
        float d2  = fmaxf(sqi + sqj - 2.0f * g, 0.0f);
        float dist = (d2 > 1e-12f) ? sqrtf(d2) : 0.0f;
        float li  = sim * dist + (1.0f - sim) * fmaxf(MARGIN - dist, 0.0f);
        lsum += w * fabsf(li);
      }
    }
  }

  lsum = waveReduceAdd(lsum);
  if (lane == 0) red[wib] = lsum;
  __syncthreads();
  if (tid == 0) {
    float s = 0.f;
    for (int i = 0; i < 8; ++i) s += red[i];
    atomicAdd(accum_hyper, s);
  }
}

// ---------------------------------------------------------------------------
// K8: loss_node = sum |x_trans - e2v_mean(he)|
// ---------------------------------------------------------------------------
__global__ void k_lossnode(const float* __restrict__ xt, const float* __restrict__ he,
                           const int* __restrict__ off, const int* __restrict__ deg,
                           const int* __restrict__ csr_e, float* __restrict__ accum_node) {
  int idx = blockIdx.x * blockDim.x + threadIdx.x;
  float local = 0.f;
  if (idx < BB * NN) {
    int n = idx & (NN - 1);
    int b = idx >> 12;
    int o = off[n];
    int cnt = deg[n];
    float inv = 1.0f / fmaxf((float)cnt, 1.0f);
    for (int c4 = 0; c4 < 16; ++c4) {
      float sx = 0.f, sy = 0.f, sz = 0.f, sw = 0.f;
      for (int s = 0; s < cnt; ++s) {
        int e = csr_e[o + s];
        const float* g = he + ((size_t)b * EE + e) * CC + c4 * 4;
        sx += g[0]; sy += g[1]; sz += g[2]; sw += g[3];
      }
      const float* x4 = xt + ((size_t)b * NN + n) * CC + c4 * 4;
      local += fabsf(x4[0] - sx * inv) + fabsf(x4[1] - sy * inv) +
               fabsf(x4[2] - sz * inv) + fabsf(x4[3] - sw * inv);
    }
  }
  local = waveReduceAdd(local);
  __shared__ float red[8];
  int wib = threadIdx.x >> 5;
  if ((threadIdx.x & 31) == 0) red[wib] = local;
  __syncthreads();
  if (threadIdx.x == 0) {
    float s = 0.f;
    int nw = blockDim.x >> 5;
    for (int i = 0; i < nw; ++i) s += red[i];
    atomicAdd(accum_node, s);
  }
}

// ---------------------------------------------------------------------------
// K9: out = relu(e2v_mean(he2))
// ---------------------------------------------------------------------------
__global__ void k_out(const float* __restrict__ he2, const int* __restrict__ off,
                      const int* __restrict__ deg, const int* __restrict__ csr_e,
                      float* __restrict__ out) {
  int idx = blockIdx.x * blockDim.x + threadIdx.x;
  if (idx >= BB * NN * 16) return;
  int c4 = idx & 15;
  int r  = idx >> 4;
  int n  = r & (NN - 1);
  int b  = r >> 12;
  int o = off[n];
  int cnt = deg[n];
  float inv = 1.0f / fmaxf((float)cnt, 1.0f);
  float sx = 0.f, sy = 0.f, sz = 0.f, sw = 0.f;
  for (int s = 0; s < cnt; ++s) {
    int e = csr_e[o + s];
    const float* g = he2 + ((size_t)b * EE + e) * CC + c4 * 4;
    sx += g[0]; sy += g[1]; sz += g[2]; sw += g[3];
  }
  float* op = out + (size_t)r * CC + c4 * 4;
  op[0] = fmaxf(sx * inv, 0.f);
  op[1] = fmaxf(sy * inv, 0.f);
  op[2] = fmaxf(sz * inv, 0.f);
  op[3] = fmaxf(sw * inv, 0.f);
}

// ---------------------------------------------------------------------------
// K10: finalize scalar loss
// ---------------------------------------------------------------------------
__global__ void k_final(const float* __restrict__ accum, float* __restrict__ out_scalar) {
  if (threadIdx.x == 0 && blockIdx.x == 0) {
    float loss_node  = accum[0] * (1.0f / (float)(BB * NN * CC));
    float loss_hyper = accum[1] * (1.0f / ((float)ROWS * (float)ROWS));
    *out_scalar = loss_node + loss_hyper;
  }
}

// ---------------------------------------------------------------------------
extern "C" void kernel_launch(void* const* d_in, const int* in_sizes, int n_in,
                              void* d_out, int out_size, void* d_ws, size_t ws_size,
                              hipStream_t stream) {
  const float* x       = (const float*)d_in[0];
  const float* W_lin   = (const float*)d_in[1];
  const float* b_lin   = (const float*)d_in[2];
  const float* W_theta = (const float*)d_in[3];
  const float* b_theta = (const float*)d_in[4];
  const int*   ids     = (const int*)d_in[5];

  float* out = (float*)d_out;

  // workspace layout
  char* base = (char*)d_ws;
  float* f_xtrans = (float*)base;                         base += (size_t)BB * NN * CC * 4;   // 8 MB
  float* f_he     = (float*)base;                         base += (size_t)ROWS * CC * 4;      // 2 MB
  float* f_he2    = (float*)base;                         base += (size_t)ROWS * CC * 4;      // 2 MB
  float* f_sq     = (float*)base;                         base += (size_t)ROWS * 4;
  float* f_accum  = (float*)base;                         base += 64;                          // [0]=node,[1]=hyper
  int*   i_deg    = (int*)base;                           base += (size_t)NN * 4;
  int*   i_off    = (int*)base;                           base += (size_t)NN * 4;
  int*   i_cur    = (int*)base;                           base += (size_t)NN * 4;
  int*   i_csre   = (int*)base;                           base += (size_t)EE * DEG * 4;

  // 0) init
  k_zero<<<(NN + 255) / 256, 256, 0, stream>>>(i_deg, f_accum);
  // 1) degrees
  k_degcount<<<(EE * DEG + 255) / 256, 256, 0, stream>>>(ids, i_deg);
  // 2) scan
  k_scan<<<1, 1, 0, stream>>>(i_deg, i_off, i_cur);
  // 3) CSR fill
  k_fill<<<(EE * DEG + 255) / 256, 256, 0, stream>>>(ids, i_cur, i_csre);
  // 4) x_trans = x @ W_lin + b_lin   (32768x64 @ 64x64) -> 8192 tile-waves
  k_gemm64<<<(BB * NN / 16 * 4 + 7) / 8, 256, 0, stream>>>(x, W_lin, b_lin, f_xtrans,
                                                            BB * NN, 1.0f);
  // 5) he gather-sum
  k_he<<<(ROWS * 16 + 255) / 256, 256, 0, stream>>>(f_xtrans, ids, f_he);
  // 6) row norms
  k_sq<<<(ROWS + 255) / 256, 256, 0, stream>>>(f_he, f_sq);
  // 7) Gram + hyper loss (dominant WMMA work, LDS-staged macro tiles)
  k_gram_loss<<<512, 256, 0, stream>>>(f_he, f_sq, f_accum + 1);
  // 8) he2 = he @ W_theta / 8 + b_theta   (8000x64 @ 64x64) -> 2000 tile-waves
  k_gemm64<<<(ROWS / 16 * 4 + 7) / 8, 256, 0, stream>>>(f_he, W_theta, b_theta, f_he2,
                                                         ROWS, 0.125f);
  // 9) node loss
  k_lossnode<<<(BB * NN + 255) / 256, 256, 0, stream>>>(f_xtrans, f_he, i_off, i_deg,
                                                         i_csre, f_accum);
  // 10) output relu(e2v_mean(he2))
  k_out<<<(BB * NN * 16 + 255) / 256, 256, 0, stream>>>(f_he2, i_off, i_deg, i_csre, out);
  // 11) scalar loss
  k_final<<<1, 1, 0, stream>>>(f_accum, out + (size_t)BB * NN * CC);

  (void)in_sizes; (void)n_in; (void)out_size; (void)ws_size;
}